// FrequencyAwareEmbedding_73796128080340
// MI455X (gfx1250) — compile-verified
//
#include <hip/hip_runtime.h>
#include <hip/hip_bf16.h>

typedef float v2f __attribute__((ext_vector_type(2)));
typedef float v8f __attribute__((ext_vector_type(8)));

#define NUM_EMB 100000
#define BASE_DIM 32

// ---------------------------------------------------------------------------
// Kernel 0: zero the two bucket counters in workspace
// ---------------------------------------------------------------------------
__global__ void fae_zero_counters(int* cnt) {
    if (threadIdx.x < 2) cnt[threadIdx.x] = 0;
}

// ---------------------------------------------------------------------------
// Kernel 1: partition tokens by bucket; buckets 0-2 are a pure gather-copy
// (handled inline), buckets 3/4 get compacted into token lists for the WMMA
// projection kernels.
// ---------------------------------------------------------------------------
__global__ void fae_partition_copy(const int* __restrict__ x,
                                   const int* __restrict__ bucket,
                                   const float* __restrict__ emb0,
                                   const float* __restrict__ emb1,
                                   const float* __restrict__ emb2,
                                   float* __restrict__ out,
                                   int* __restrict__ cnt,      // [2]
                                   int* __restrict__ list3,
                                   int* __restrict__ list4,
                                   int tok_base, int tok_count)
{
    int t = blockIdx.x * blockDim.x + threadIdx.x;
    if (t >= tok_count) return;
    int tok = tok_base + t;
    int idx = x[tok];
    int bk  = bucket[idx];
    if (bk < 3) {
        const float* tab = (bk == 0) ? emb0 : (bk == 1) ? emb1 : emb2;
        const float4* src = (const float4*)(tab + (size_t)idx * BASE_DIM);
        float4* dst = (float4*)(out + (size_t)tok * BASE_DIM);
#pragma unroll
        for (int j = 0; j < 8; ++j) dst[j] = src[j];
    } else if (bk == 3) {
        int p = atomicAdd(&cnt[0], 1);
        list3[p] = tok;
    } else {
        int p = atomicAdd(&cnt[1], 1);
        list4[p] = tok;
    }
}

// ---------------------------------------------------------------------------
// Kernel 2: WMMA projection for compacted bucket-3/4 tokens.
// Each wave (32 lanes) owns 16 tokens:
//   A tile: 16 x KP gathered embedding rows (f32), consumed 16x4 per WMMA
//   B tile: W[KP x 32], two 16-wide N tiles
//   D = A*W + bias, scattered back to out[token*32 + n]
// f32 16x16x4 layouts (ISA 7.12.2):
//   A: lanes 0-15 hold M=0..15 {K=k,k+1}; lanes 16-31 hold {K=k+2,k+3}
//   B: lanes 0-15 hold N=0..15 {K=k,k+1}; lanes 16-31 hold {K=k+2,k+3}
//   D: VGPR v -> row (lane<16 ? v : v+8), col = lane&15
// ---------------------------------------------------------------------------
template <int KD>
__global__ void fae_project_wmma(const int* __restrict__ x,
                                 const float* __restrict__ emb,   // [NUM_EMB, KD]
                                 const float* __restrict__ W,     // [KD, 32]
                                 const float* __restrict__ bias,  // [32]
                                 const int* __restrict__ list,
                                 const int* __restrict__ cnt_ptr,
                                 float* __restrict__ out)
{
    const int count = *cnt_ptr;
    const int wave  = (int)((blockIdx.x * blockDim.x + threadIdx.x) >> 5);
    const int lane  = (int)(threadIdx.x & 31);
    const int base  = wave * 16;
    if (base >= count) return;           // wave-uniform exit

    const int half = lane >> 4;          // 0: K pair {k,k+1}, 1: {k+2,k+3}
    const int m    = lane & 15;          // A row index / B-D column index

    // Gather this lane's token row (clamped for the tail; tail rows are
    // computed but never stored).
    int ti = base + m;
    int tc = (ti < count) ? ti : (count - 1);
    int tok = list[tc];
    const float* row = emb + (size_t)x[tok] * KD;

    v8f acc0 = {};   // N = 0..15
    v8f acc1 = {};   // N = 16..31
    constexpr int KP = (KD + 3) & ~3;

#pragma unroll
    for (int k = 0; k < KP; k += 4) {
        const int kk = k + half * 2;
        v2f a, b0, b1;
        a.x  = (kk     < KD) ? row[kk]     : 0.0f;
        a.y  = (kk + 1 < KD) ? row[kk + 1] : 0.0f;
        b0.x = (kk     < KD) ? W[kk * BASE_DIM + m]            : 0.0f;
        b0.y = (kk + 1 < KD) ? W[(kk + 1) * BASE_DIM + m]      : 0.0f;
        b1.x = (kk     < KD) ? W[kk * BASE_DIM + m + 16]       : 0.0f;
        b1.y = (kk + 1 < KD) ? W[(kk + 1) * BASE_DIM + m + 16] : 0.0f;
        acc0 = __builtin_amdgcn_wmma_f32_16x16x4_f32(false, a, false, b0,
                                                     (short)0, acc0, false, false);
        acc1 = __builtin_amdgcn_wmma_f32_16x16x4_f32(false, a, false, b1,
                                                     (short)0, acc1, false, false);
    }

    const float bia0 = bias[m];
    const float bia1 = bias[m + 16];

#pragma unroll
    for (int v = 0; v < 8; ++v) {
        int rrow = half ? (v + 8) : v;
        int ri = base + rrow;
        if (ri < count) {                 // lane-predicated store only
            int tok_r = list[ri];
            float* o = out + (size_t)tok_r * BASE_DIM;
            o[m]      = acc0[v] + bia0;
            o[m + 16] = acc1[v] + bia1;
        }
    }
}

// ---------------------------------------------------------------------------
// Fallback (only used if the workspace is unusably small): direct per-token
// VALU path, no compaction.
// ---------------------------------------------------------------------------
__global__ void fae_direct_all(const int* __restrict__ x,
                               const int* __restrict__ bucket,
                               const float* __restrict__ emb0,
                               const float* __restrict__ emb1,
                               const float* __restrict__ emb2,
                               const float* __restrict__ emb3,
                               const float* __restrict__ emb4,
                               const float* __restrict__ W3,
                               const float* __restrict__ b3,
                               const float* __restrict__ W4,
                               const float* __restrict__ b4,
                               float* __restrict__ out, int T)
{
    int t = blockIdx.x * blockDim.x + threadIdx.x;
    if (t >= T) return;
    int idx = x[t];
    int bk  = bucket[idx];
    float* o = out + (size_t)t * BASE_DIM;
    if (bk < 3) {
        const float* s = ((bk == 0) ? emb0 : (bk == 1) ? emb1 : emb2)
                         + (size_t)idx * BASE_DIM;
#pragma unroll
        for (int j = 0; j < BASE_DIM; ++j) o[j] = s[j];
    } else {
        const float* row; const float* W; const float* b; int KD;
        if (bk == 3) { row = emb3 + (size_t)idx * 51;  W = W3; b = b3; KD = 51; }
        else         { row = emb4 + (size_t)idx * 102; W = W4; b = b4; KD = 102; }
        for (int n = 0; n < BASE_DIM; ++n) {
            float acc = b[n];
            for (int k = 0; k < KD; ++k) acc += row[k] * W[k * BASE_DIM + n];
            o[n] = acc;
        }
    }
}

// ---------------------------------------------------------------------------
// Host launch
// ---------------------------------------------------------------------------
extern "C" void kernel_launch(void* const* d_in, const int* in_sizes, int n_in,
                              void* d_out, int out_size, void* d_ws, size_t ws_size,
                              hipStream_t stream)
{
    const int*   x      = (const int*)  d_in[0];
    const int*   bucket = (const int*)  d_in[1];
    const float* emb0   = (const float*)d_in[2];
    const float* emb1   = (const float*)d_in[3];
    const float* emb2   = (const float*)d_in[4];
    const float* emb3   = (const float*)d_in[5];
    const float* emb4   = (const float*)d_in[6];
    const float* W3     = (const float*)d_in[7];
    const float* b3     = (const float*)d_in[8];
    const float* W4     = (const float*)d_in[9];
    const float* b4     = (const float*)d_in[10];
    float* out = (float*)d_out;

    const long long T = in_sizes[0];   // BATCH*SEQ tokens

    // Workspace layout: [0..1] counters, then two token lists of `chunk` ints.
    long long avail = 0;
    if (ws_size > 32) avail = (long long)((ws_size - 32) / (2 * sizeof(int)));
    if (avail < 1024) {
        // Unusable workspace: direct VALU fallback.
        int blocks = (int)((T + 255) / 256);
        fae_direct_all<<<blocks, 256, 0, stream>>>(x, bucket, emb0, emb1, emb2,
                                                   emb3, emb4, W3, b3, W4, b4,
                                                   out, (int)T);
        return;
    }

    long long chunk = (avail < T) ? avail : T;
    int* cnt   = (int*)d_ws;
    int* list3 = cnt + 8;                      // 32-byte aligned
    int* list4 = list3 + chunk;

    for (long long start = 0; start < T; start += chunk) {
        long long c = T - start;
        if (c > chunk) c = chunk;

        fae_zero_counters<<<1, 32, 0, stream>>>(cnt);

        int pblocks = (int)((c + 255) / 256);
        fae_partition_copy<<<pblocks, 256, 0, stream>>>(
            x, bucket, emb0, emb1, emb2, out, cnt, list3, list4,
            (int)start, (int)c);

        // Worst case: all c tokens in one bucket -> ceil(c/16) waves.
        long long groups  = (c + 15) / 16;
        long long threads = groups * 32;
        int wblocks = (int)((threads + 255) / 256);

        fae_project_wmma<51><<<wblocks, 256, 0, stream>>>(
            x, emb3, W3, b3, list3, &cnt[0], out);
        fae_project_wmma<102><<<wblocks, 256, 0, stream>>>(
            x, emb4, W4, b4, list4, &cnt[1], out);
    }
}